// GATGraphLabeller_22058952032416
// MI455X (gfx1250) — compile-verified
//
#include <hip/hip_runtime.h>
#include <float.h>
#include <math.h>

#define NN 100000
#define NE 1600000
#define H  64
#define NG 1024
#define NT 10
#define SLOPE 0.2f

typedef __attribute__((ext_vector_type(2))) float v2f;
typedef __attribute__((ext_vector_type(8))) float v8f;

// ---------------------------------------------------------------- init
__global__ __launch_bounds__(256) void k_init(float* __restrict__ agg,
                                              float* __restrict__ den,
                                              float* __restrict__ mmax,
                                              float* __restrict__ sums,
                                              float* __restrict__ cnt) {
    int i = blockIdx.x * 256 + threadIdx.x;
    if (i < NN * H) agg[i] = 0.f;
    if (i < NN)     { den[i] = 0.f; mmax[i] = -FLT_MAX; }
    if (i < NG * H) sums[i] = 0.f;
    if (i < NG)     cnt[i] = 0.f;
}

// ------------------------------------------------- fold attention vectors
// vsrc = W_gat @ att_src, vdst = W_gat @ att_dst,
// we   = W_edge @ att_edge (67) -> tab[t] = emb_travel[t] . we[0:64], wcont = we[64:67]
__global__ __launch_bounds__(256) void k_setup(const float* __restrict__ Wg,
                                               const float* __restrict__ att_src,
                                               const float* __restrict__ att_dst,
                                               const float* __restrict__ W_edge,
                                               const float* __restrict__ att_edge,
                                               const float* __restrict__ emb_trav,
                                               float* __restrict__ vsrc,
                                               float* __restrict__ vdst,
                                               float* __restrict__ tab,
                                               float* __restrict__ wcont) {
    __shared__ float s_we[H + 3];
    const int i = threadIdx.x;
    if (i < H) {
        float a = 0.f;
        for (int n = 0; n < H; ++n) a = fmaf(Wg[i * H + n], att_src[n], a);
        vsrc[i] = a;
    } else if (i < 2 * H) {
        const int k = i - H;
        float a = 0.f;
        for (int n = 0; n < H; ++n) a = fmaf(Wg[k * H + n], att_dst[n], a);
        vdst[k] = a;
    } else if (i < 2 * H + H + 3) {
        const int c = i - 2 * H;               // 0..66
        float a = 0.f;
        for (int n = 0; n < H; ++n) a = fmaf(W_edge[c * H + n], att_edge[n], a);
        s_we[c] = a;
    }
    __syncthreads();
    if (i < 8) {
        float a = 0.f;
        for (int c = 0; c < H; ++c) a = fmaf(emb_trav[i * H + c], s_we[c], a);
        tab[i] = a;
    }
    if (i < 3) wcont[i] = s_we[H + i];
}

// ---------------------------------------- node kernel: x, h = x@W (WMMA f32)
__global__ __launch_bounds__(256) void k_node(const int* __restrict__ act,
                                              const int* __restrict__ loc,
                                              const float* __restrict__ emb_act,
                                              const float* __restrict__ emb_loc,
                                              const float* __restrict__ Wg,
                                              const float* __restrict__ vsrc,
                                              const float* __restrict__ vdst,
                                              float* __restrict__ h,
                                              float* __restrict__ asrc,
                                              float* __restrict__ adst) {
    constexpr int WS = 65;                      // padded LDS stride (bank-conflict free)
    __shared__ float sW[64 * WS];
    __shared__ float sX[8 * 16 * WS];
    __shared__ float sV[2 * H];
    const int tid = threadIdx.x;
    for (int i = tid; i < 64 * 64; i += 256)
        sW[(i >> 6) * WS + (i & 63)] = Wg[i];
    if (tid < H)          sV[tid] = vsrc[tid];
    else if (tid < 2 * H) sV[tid] = vdst[tid - H];
    __syncthreads();

    const int wave = tid >> 5;
    const int lane = tid & 31;
    const int m0 = (blockIdx.x * 8 + wave) * 16;
    if (m0 >= NN) return;

    float* xt = &sX[wave * 16 * WS];
    const int half = lane >> 4;                 // 0: lanes 0-15, 1: lanes 16-31
    const int rrow = lane & 15;
    const int c0 = half * 32;

    // build relu(emb_act + emb_loc) tile: 2 lanes per row, 32 cols each
    {
        int node = m0 + rrow;
        if (node >= NN) node = NN - 1;
        const float* ea = &emb_act[act[node] * H];
        const float* el = &emb_loc[loc[node] * H];
        #pragma unroll
        for (int j = 0; j < 32; ++j) {
            float v = ea[c0 + j] + el[c0 + j];
            xt[rrow * WS + c0 + j] = v > 0.f ? v : 0.f;
        }
    }
    // fused a_src / a_dst = x . (W@att)  (half-wave shuffle reduction)
    {
        float ps = 0.f, pd = 0.f;
        #pragma unroll
        for (int j = 0; j < 32; ++j) {
            const float xv = xt[rrow * WS + c0 + j];
            ps = fmaf(xv, sV[c0 + j], ps);
            pd = fmaf(xv, sV[H + c0 + j], pd);
        }
        ps += __shfl_xor(ps, 16, 32);
        pd += __shfl_xor(pd, 16, 32);
        if (lane < 16 && (m0 + rrow) < NN) {
            asrc[m0 + rrow] = ps;
            adst[m0 + rrow] = pd;
        }
    }
    // h = x @ W_gat : 16x64 out tile = 4 N-tiles, K=64 in 16 steps of fp32 WMMA
    v8f acc0 = {}, acc1 = {}, acc2 = {}, acc3 = {};
    const int koff = half * 2;                  // A/B VGPR0 holds K=0 | K=2 per half
    const int nlo = lane & 15;
    #pragma unroll
    for (int ks = 0; ks < 16; ++ks) {
        const int kk = ks * 4 + koff;
        v2f a;
        a.x = xt[rrow * WS + kk];
        a.y = xt[rrow * WS + kk + 1];
        v2f b0, b1, b2, b3;
        b0.x = sW[kk * WS + nlo];        b0.y = sW[(kk + 1) * WS + nlo];
        b1.x = sW[kk * WS + 16 + nlo];   b1.y = sW[(kk + 1) * WS + 16 + nlo];
        b2.x = sW[kk * WS + 32 + nlo];   b2.y = sW[(kk + 1) * WS + 32 + nlo];
        b3.x = sW[kk * WS + 48 + nlo];   b3.y = sW[(kk + 1) * WS + 48 + nlo];
        acc0 = __builtin_amdgcn_wmma_f32_16x16x4_f32(false, a, false, b0, (short)0, acc0, false, false);
        acc1 = __builtin_amdgcn_wmma_f32_16x16x4_f32(false, a, false, b1, (short)0, acc1, false, false);
        acc2 = __builtin_amdgcn_wmma_f32_16x16x4_f32(false, a, false, b2, (short)0, acc2, false, false);
        acc3 = __builtin_amdgcn_wmma_f32_16x16x4_f32(false, a, false, b3, (short)0, acc3, false, false);
    }
    // C/D layout: VGPR r -> rows m0+r (lanes 0-15) / m0+r+8 (lanes 16-31), col = nlo
    #pragma unroll
    for (int r = 0; r < 8; ++r) {
        const int row = m0 + r + half * 8;
        if (row < NN) {
            float* hp = &h[row * H + nlo];
            hp[0]  = acc0[r];
            hp[16] = acc1[r];
            hp[32] = acc2[r];
            hp[48] = acc3[r];
        }
    }
}

// ------------------------------------------- edge pass A: alpha + segment max
__global__ __launch_bounds__(256) void k_alpha(const int* __restrict__ ei,
                                               const int* __restrict__ travel,
                                               const float* __restrict__ dur,
                                               const float* __restrict__ tst,
                                               const float* __restrict__ tet,
                                               const float* __restrict__ asrc,
                                               const float* __restrict__ adst,
                                               const float* __restrict__ tab,
                                               const float* __restrict__ wcont,
                                               float* __restrict__ alpha,
                                               float* __restrict__ mmax) {
    const int e = blockIdx.x * 256 + threadIdx.x;
    if (e >= NE) return;
    const int s = ei[e];
    const int d = ei[NE + e];
    float v = asrc[s] + adst[d] + tab[travel[e]];
    v = fmaf(wcont[0], dur[e], v);
    v = fmaf(wcont[1], tst[e], v);
    v = fmaf(wcont[2], tet[e], v);
    v = v > 0.f ? v : SLOPE * v;                // leaky_relu
    alpha[e] = v;
    // float atomic-max via signed/unsigned trick (init = -FLT_MAX)
    if (v >= 0.f) atomicMax((int*)&mmax[d], __float_as_int(v));
    else          atomicMin((unsigned int*)&mmax[d], __float_as_uint(v));
}

// ---------------------------------------- edge pass B: exp + segment sum
__global__ __launch_bounds__(256) void k_exp(const int* __restrict__ ei,
                                             const float* __restrict__ mmax,
                                             float* __restrict__ alpha,
                                             float* __restrict__ den) {
    const int e = blockIdx.x * 256 + threadIdx.x;
    if (e >= NE) return;
    const int d = ei[NE + e];
    const float ex = expf(alpha[e] - mmax[d]);
    alpha[e] = ex;                               // reuse buffer: now holds exp
    atomicAdd(&den[d], ex);
}

// ------------------------- edge pass C: out[dst] += coef * h[src] (wave/edge)
__global__ __launch_bounds__(256) void k_agg(const int* __restrict__ ei,
                                             const float* __restrict__ alpha,
                                             const float* __restrict__ den,
                                             const float* __restrict__ h,
                                             float* __restrict__ agg) {
    const int gt = blockIdx.x * 256 + threadIdx.x;
    const int e = gt >> 5;
    if (e >= NE) return;
    const int lane = gt & 31;
    const int s = ei[e];
    const int d = ei[NE + e];
    const float coef = alpha[e] / fmaxf(den[d], 1e-16f);
    const float* hs = &h[s * H];
    float* op = &agg[d * H];
    atomicAdd(&op[lane],      coef * hs[lane]);
    atomicAdd(&op[lane + 32], coef * hs[lane + 32]);
}

// --------------------------------------- relu(+bias) + graph mean-pool sums
__global__ __launch_bounds__(64) void k_pool(const float* __restrict__ agg,
                                             const float* __restrict__ b_gat,
                                             const int* __restrict__ batch,
                                             float* __restrict__ sums,
                                             float* __restrict__ cnt) {
    const int n = blockIdx.x;
    const int c = threadIdx.x;
    float v = agg[n * H + c] + b_gat[c];
    v = v > 0.f ? v : 0.f;
    const int b = batch[n];
    atomicAdd(&sums[b * H + c], v);
    if (c == 0) atomicAdd(&cnt[b], 1.0f);
}

// ------------------------------------------------ FC head + log_softmax
__global__ __launch_bounds__(256) void k_head(const float* __restrict__ sums,
                                              const float* __restrict__ cnt,
                                              const float* __restrict__ W_fc,
                                              const float* __restrict__ b_fc,
                                              float* __restrict__ out) {
    const int g = blockIdx.x * 256 + threadIdx.x;
    if (g >= NG) return;
    const float inv = 1.0f / fmaxf(cnt[g], 1.0f);
    float logits[NT];
    #pragma unroll
    for (int t = 0; t < NT; ++t) logits[t] = b_fc[t];
    for (int c = 0; c < H; ++c) {
        const float p = sums[g * H + c] * inv;
        #pragma unroll
        for (int t = 0; t < NT; ++t) logits[t] = fmaf(p, W_fc[c * NT + t], logits[t]);
    }
    float mx = logits[0];
    #pragma unroll
    for (int t = 1; t < NT; ++t) mx = fmaxf(mx, logits[t]);
    float se = 0.f;
    #pragma unroll
    for (int t = 0; t < NT; ++t) se += expf(logits[t] - mx);
    const float lse = mx + logf(se);
    #pragma unroll
    for (int t = 0; t < NT; ++t) out[g * NT + t] = logits[t] - lse;
}

// ================================================================= launch
extern "C" void kernel_launch(void* const* d_in, const int* in_sizes, int n_in,
                              void* d_out, int out_size, void* d_ws, size_t ws_size,
                              hipStream_t stream) {
    const int*   act      = (const int*)d_in[0];
    const int*   loc      = (const int*)d_in[1];
    const int*   travel   = (const int*)d_in[2];
    const float* dur      = (const float*)d_in[3];
    const float* tst      = (const float*)d_in[4];
    const float* tet      = (const float*)d_in[5];
    const int*   ei       = (const int*)d_in[6];
    const int*   batch    = (const int*)d_in[7];
    const float* emb_act  = (const float*)d_in[8];
    const float* emb_loc  = (const float*)d_in[9];
    const float* emb_trav = (const float*)d_in[10];
    const float* Wg       = (const float*)d_in[11];
    const float* att_src  = (const float*)d_in[12];
    const float* att_dst  = (const float*)d_in[13];
    const float* W_edge   = (const float*)d_in[14];
    const float* att_edge = (const float*)d_in[15];
    const float* b_gat    = (const float*)d_in[16];
    const float* W_fc     = (const float*)d_in[17];
    const float* b_fc     = (const float*)d_in[18];
    float* out = (float*)d_out;

    float* ws = (float*)d_ws;
    size_t o = 0;
    float* h     = ws + o; o += (size_t)NN * H;   // 6.4M
    float* asrc  = ws + o; o += NN;
    float* adst  = ws + o; o += NN;
    float* alpha = ws + o; o += NE;               // 1.6M (alpha -> exp in place)
    float* mmax  = ws + o; o += NN;
    float* den   = ws + o; o += NN;
    float* agg   = ws + o; o += (size_t)NN * H;   // 6.4M
    float* sums  = ws + o; o += (size_t)NG * H;
    float* cnt   = ws + o; o += NG;
    float* vsrc  = ws + o; o += H;
    float* vdst  = ws + o; o += H;
    float* tab   = ws + o; o += 8;
    float* wcont = ws + o; o += 3;

    k_init <<<(NN * H + 255) / 256, 256, 0, stream>>>(agg, den, mmax, sums, cnt);
    k_setup<<<1, 256, 0, stream>>>(Wg, att_src, att_dst, W_edge, att_edge, emb_trav,
                                   vsrc, vdst, tab, wcont);
    k_node <<<((NN + 15) / 16 + 7) / 8, 256, 0, stream>>>(act, loc, emb_act, emb_loc,
                                                          Wg, vsrc, vdst, h, asrc, adst);
    k_alpha<<<(NE + 255) / 256, 256, 0, stream>>>(ei, travel, dur, tst, tet,
                                                  asrc, adst, tab, wcont, alpha, mmax);
    k_exp  <<<(NE + 255) / 256, 256, 0, stream>>>(ei, mmax, alpha, den);
    k_agg  <<<(NE * 32) / 256, 256, 0, stream>>>(ei, alpha, den, h, agg);
    k_pool <<<NN, 64, 0, stream>>>(agg, b_gat, batch, sums, cnt);
    k_head <<<(NG + 255) / 256, 256, 0, stream>>>(sums, cnt, W_fc, b_fc, out);
}